// DigitCap_53412213293297
// MI455X (gfx1250) — compile-verified
//
#include <hip/hip_runtime.h>

// DigitCaps routing for MI455X (gfx1250, wave32, WMMA).
// s-GEMM per (b_tile, d): [16 x 9216] x [9216 x 16] via V_WMMA_F32_16X16X4_F32.
// u_hat (189 MB) is never materialized; workspace is ~24 MB (L2-resident).

#define B_SZ 256
#define ND   10
#define NP   1152
#define DI   16
#define DJ   8
#define KTOT (NP * DJ)   // 9216

typedef float v2f __attribute__((ext_vector_type(2)));
typedef float v8f __attribute__((ext_vector_type(8)));

// ---------------------------------------------------------------------------
// c = softmax(logits, axis=d).  One thread per (b, p); 10 coalesced passes.
// ---------------------------------------------------------------------------
__global__ __launch_bounds__(256)
void softmax_c_kernel(const float* __restrict__ logits, float* __restrict__ c)
{
    int t = blockIdx.x * 256 + threadIdx.x;        // over B*NP (exact)
    int b = t / NP, p = t % NP;
    const float* lp = logits + (size_t)b * ND * NP + p;
    float l[ND];
    float mx = -1e30f;
#pragma unroll
    for (int d = 0; d < ND; ++d) { l[d] = lp[(size_t)d * NP]; mx = fmaxf(mx, l[d]); }
    float sum = 0.0f;
#pragma unroll
    for (int d = 0; d < ND; ++d) { l[d] = __expf(l[d] - mx); sum += l[d]; }
    float inv = 1.0f / sum;
    float* cp = c + (size_t)b * ND * NP + p;
#pragma unroll
    for (int d = 0; d < ND; ++d) cp[(size_t)d * NP] = l[d] * inv;
}

// ---------------------------------------------------------------------------
// s[b,d,:] = sum_k (c*pc)[b,k] * W'[k,:]  — WMMA f32 16x16x4, K = 9216.
// Grid: (16 b-tiles, 10 digits).  8 waves split K; LDS ds_add_f32 reduce.
// K-loop steps by 8 (= one primary capsule p): p = k0>>3 is wave-uniform,
// one c-load feeds two WMMAs, W/pc addresses fold to immediate offsets.
// Templated on UNIFORM_C so round 0 (softmax(0) = 0.1) has a branch-free loop.
// Also emits per-block partial of n2 = sum(s^2) for the global squash norm.
// ---------------------------------------------------------------------------
template <bool UNIFORM_C>
__global__ __launch_bounds__(256)
void s_gemm_kernel(const float* __restrict__ pc,      // [B][NP][DJ]
                   const float* __restrict__ W,       // [ND][NP][DI][DJ]
                   const float* __restrict__ c,       // [B][ND][NP]
                   float* __restrict__ s_out,         // [B][ND][DI]
                   float* __restrict__ n2_partial)    // [160]
{
    __shared__ float s_acc[16 * DI];   // [bb][i]
    __shared__ float red[256];

    const int tid   = threadIdx.x;
    const int lane  = tid & 31;
    const int wave  = tid >> 5;        // 0..7
    const int btile = blockIdx.x;      // 0..15
    const int d     = blockIdx.y;      // 0..9
    const int m     = lane & 15;       // A: row bb  /  B: col i
    const int h     = lane >> 4;       // half select (K offset +2)

    s_acc[tid] = 0.0f;
    __syncthreads();

    const int    b   = btile * 16 + m;
    const float* pcb = pc + (size_t)b * KTOT;
    const float* cb  = c  + ((size_t)b * ND + d) * NP;
    const float* Wd  = W  + (size_t)d * NP * DI * DJ;

    v8f acc = {};
    const int k_begin = wave * (KTOT / 8);          // contiguous K chunk
    const int k_end   = k_begin + (KTOT / 8);
    for (int k0 = k_begin; k0 < k_end; k0 += 8) {   // one p per iteration
        const int   p  = k0 >> 3;                   // wave-uniform
        const float cv = UNIFORM_C ? 0.1f : cb[p];
        const float* pcp = pcb + (k0 + 2 * h);      // A rows for this lane
        const float* wp  = Wd + ((size_t)p * DI + m) * DJ + 2 * h;

        v2f a0 = { pcp[0] * cv, pcp[1] * cv };      // K = k0+2h, k0+2h+1
        v2f b0 = { wp[0],  wp[1]  };                // W[d,p,m, 2h], [.., 2h+1]
        acc = __builtin_amdgcn_wmma_f32_16x16x4_f32(
            false, a0, false, b0, (short)0, acc, false, false);

        v2f a1 = { pcp[4] * cv, pcp[5] * cv };      // K = k0+4+2h, k0+5+2h
        v2f b1 = { wp[4],  wp[5]  };                // W[d,p,m, 4+2h], [.., 5+2h]
        acc = __builtin_amdgcn_wmma_f32_16x16x4_f32(
            false, a1, false, b1, (short)0, acc, false, false);
    }

    // D layout: lane holds s[bb = r + 8h, i = m] in acc[r]
#pragma unroll
    for (int r = 0; r < 8; ++r)
        atomicAdd(&s_acc[(r + 8 * h) * DI + m], acc[r]);
    __syncthreads();

    // tid == bb*16 + i : write s, reduce n2 partial
    float val  = s_acc[tid];
    int   bb   = tid >> 4;
    int   i    = tid & 15;
    int   bout = btile * 16 + bb;
    s_out[((size_t)bout * ND + d) * DI + i] = val;

    red[tid] = val * val;
    __syncthreads();
    for (int off = 128; off > 0; off >>= 1) {
        if (tid < off) red[tid] += red[tid + off];
        __syncthreads();
    }
    if (tid == 0) n2_partial[blockIdx.y * 16 + blockIdx.x] = red[0];
}

// ---------------------------------------------------------------------------
// v = squash(s) with GLOBAL Frobenius norm (faithful to reference).
// Each block redundantly reduces the 160 partials (uniform -> scalar loads),
// scales 256 elements.  Grid: 160 blocks cover B*ND*DI = 40960 exactly.
// ---------------------------------------------------------------------------
__global__ __launch_bounds__(256)
void squash_kernel(const float* __restrict__ s, const float* __restrict__ part,
                   float* __restrict__ v)
{
    float n2 = 0.0f;
    for (int i = 0; i < 160; ++i) n2 += part[i];
    float n     = sqrtf(n2);
    float scale = (n2 / (n2 + 1.0f)) / (n + 1e-7f);
    int idx = blockIdx.x * 256 + threadIdx.x;
    v[idx] = s[idx] * scale;
}

// ---------------------------------------------------------------------------
// logits[b,d,p] (+)= agree = sum_j pc[b,p,j] * (sum_i W[d,p,i,j] * v[b,d,i]).
// One thread per (b,d,p); t maps directly onto logits' flat layout.
// ---------------------------------------------------------------------------
__global__ __launch_bounds__(256)
void agree_kernel(const float* __restrict__ pc, const float* __restrict__ W,
                  const float* __restrict__ v, float* __restrict__ logits,
                  int first_round)
{
    int t  = blockIdx.x * 256 + threadIdx.x;       // over B*ND*NP (exact)
    int p  = t % NP;
    int bd = t / NP;
    int d  = bd % ND;

    const float* vv  = v  + (size_t)bd * DI;
    const float* wp  = W  + ((size_t)d * NP + p) * DI * DJ;
    const float* pcp = pc + ((size_t)(bd / ND) * NP + p) * DJ;

    float g[DJ] = {0, 0, 0, 0, 0, 0, 0, 0};
#pragma unroll
    for (int i = 0; i < DI; ++i) {
        float vi = vv[i];
#pragma unroll
        for (int j = 0; j < DJ; ++j) g[j] = fmaf(vi, wp[i * DJ + j], g[j]);
    }
    float agree = 0.0f;
#pragma unroll
    for (int j = 0; j < DJ; ++j) agree = fmaf(g[j], pcp[j], agree);

    if (first_round) logits[t] = agree;            // b starts at 0
    else             logits[t] += agree;
}

// ---------------------------------------------------------------------------
extern "C" void kernel_launch(void* const* d_in, const int* in_sizes, int n_in,
                              void* d_out, int out_size, void* d_ws, size_t ws_size,
                              hipStream_t stream)
{
    (void)in_sizes; (void)n_in; (void)out_size; (void)ws_size;
    const float* pc = (const float*)d_in[0];   // [256,1152,8]
    const float* W  = (const float*)d_in[1];   // [10,1152,16,8]
    float* out = (float*)d_out;                // [256,10,16]

    // Workspace layout (~24 MB, all f32):
    float* logits = (float*)d_ws;                          // B*ND*NP
    float* c      = logits + (size_t)B_SZ * ND * NP;       // B*ND*NP
    float* s      = c      + (size_t)B_SZ * ND * NP;       // B*ND*DI
    float* v      = s      + (size_t)B_SZ * ND * DI;       // B*ND*DI
    float* part   = v      + (size_t)B_SZ * ND * DI;       // 160

    const dim3 gs(16, ND);                                 // (b-tile, d)
    const int  nAgree = (B_SZ * ND * NP) / 256;            // 11520
    const int  nSoft  = (B_SZ * NP) / 256;                 // 1152
    const int  nSq    = (B_SZ * ND * DI) / 256;            // 160

    // ---- round 0 (c uniform = 0.1; logits written, never read before) ----
    s_gemm_kernel<true><<<gs, 256, 0, stream>>>(pc, W, c, s, part);
    squash_kernel<<<nSq, 256, 0, stream>>>(s, part, v);
    agree_kernel<<<nAgree, 256, 0, stream>>>(pc, W, v, logits, 1);

    // ---- round 1 ----
    softmax_c_kernel<<<nSoft, 256, 0, stream>>>(logits, c);
    s_gemm_kernel<false><<<gs, 256, 0, stream>>>(pc, W, c, s, part);
    squash_kernel<<<nSq, 256, 0, stream>>>(s, part, v);
    agree_kernel<<<nAgree, 256, 0, stream>>>(pc, W, v, logits, 0);

    // ---- round 2 (final v goes straight to d_out) ----
    softmax_c_kernel<<<nSoft, 256, 0, stream>>>(logits, c);
    s_gemm_kernel<false><<<gs, 256, 0, stream>>>(pc, W, c, s, part);
    squash_kernel<<<nSq, 256, 0, stream>>>(s, part, out);
}